// RNNet_40965398069466
// MI455X (gfx1250) — compile-verified
//
#include <hip/hip_runtime.h>

typedef __attribute__((ext_vector_type(2))) float v2f;
typedef __attribute__((ext_vector_type(4))) float v4f;
typedef __attribute__((ext_vector_type(8))) float v8f;

#define T_STEPS 32
#define B_DIM   64
#define I_DIM   256
#define H_DIM   512
#define M_DIM   (T_STEPS * B_DIM)   // 2048 rows (t*B + b)
#define BETA    0.9f
#define THRESH  1.0f

// ---------------------------------------------------------------------------
// Kernel 1: cur[r,h] = sum_k x[r,k] * W1[h,k] + b1[h]
//   C = A * B with A = x (2048x256 row-major), B = W1^T (256x512; W1 row-major)
//   V_WMMA_F32_16X16X4_F32, wave32. Per the CDNA5 ISA 32-bit layouts:
//     A 16x4 : lane holds M = lane&15, elems (K = 2*(lane>>4) + v), v=0,1
//     B 4x16 : lane holds N = lane&15, elems (K = 2*(lane>>4) + v), v=0,1
//   -> both operands are a float2 load at [row][k + 2*hi] from row-major K-
//   contiguous storage. Each wave computes a 16(M) x 64(N) strip: A loaded
//   once per K-step, reused across 4 accumulators.
// ---------------------------------------------------------------------------
__global__ __launch_bounds__(128)
void gemm_cur_kernel(const float* __restrict__ x,
                     const float* __restrict__ W1,
                     const float* __restrict__ b1,
                     float* __restrict__ cur)
{
    const int lane    = threadIdx.x & 31;
    const int wave    = threadIdx.x >> 5;     // 0..3 -> M sub-tile
    const int lanelow = lane & 15;
    const int hi      = lane >> 4;            // K half select

    const int rbase = blockIdx.x * 64 + wave * 16;   // M tile base
    const int hbase = blockIdx.y * 64;               // N strip base

    const float* aptr = x  + (size_t)(rbase + lanelow) * I_DIM + 2 * hi;
    const float* bp0  = W1 + (size_t)(hbase +  0 + lanelow) * I_DIM + 2 * hi;
    const float* bp1  = W1 + (size_t)(hbase + 16 + lanelow) * I_DIM + 2 * hi;
    const float* bp2  = W1 + (size_t)(hbase + 32 + lanelow) * I_DIM + 2 * hi;
    const float* bp3  = W1 + (size_t)(hbase + 48 + lanelow) * I_DIM + 2 * hi;

    v8f acc0 = {}, acc1 = {}, acc2 = {}, acc3 = {};

    #pragma unroll 4
    for (int k = 0; k < I_DIM; k += 4) {
        v2f a  = *(const v2f*)(aptr + k);
        v2f b0 = *(const v2f*)(bp0  + k);
        v2f b1v = *(const v2f*)(bp1 + k);
        v2f b2 = *(const v2f*)(bp2  + k);
        v2f b3 = *(const v2f*)(bp3  + k);
        acc0 = __builtin_amdgcn_wmma_f32_16x16x4_f32(false, a, false, b0,
                                                     (short)0, acc0, false, false);
        acc1 = __builtin_amdgcn_wmma_f32_16x16x4_f32(false, a, false, b1v,
                                                     (short)0, acc1, false, false);
        acc2 = __builtin_amdgcn_wmma_f32_16x16x4_f32(false, a, false, b2,
                                                     (short)0, acc2, false, false);
        acc3 = __builtin_amdgcn_wmma_f32_16x16x4_f32(false, a, false, b3,
                                                     (short)0, acc3, false, false);
    }

    // Epilogue: C layout -> lane holds N = lane&15; VGPR p holds M = p + 8*hi.
    v8f accs[4] = {acc0, acc1, acc2, acc3};
    #pragma unroll
    for (int j = 0; j < 4; ++j) {
        const int h    = hbase + j * 16 + lanelow;
        const float bb = b1[h];
        #pragma unroll
        for (int p = 0; p < 8; ++p) {
            const int r = rbase + p + 8 * hi;
            cur[(size_t)r * H_DIM + h] = accs[j][p] + bb;
        }
    }
}

// ---------------------------------------------------------------------------
// Kernel 2: leaky-integrate-and-fire scan over 32 steps. cur is constant per
// step, so each (r,h) element is an independent recurrence:
//   reset   = (mem > 1)               (from PREVIOUS mem)
//   mem     = 0.9*mem + cur - reset
//   spk[s]  = (mem > 1) ? 1 : 0
// Each thread handles 4 consecutive h values -> one coalesced 128-bit store
// per step. Store traffic (128 MiB) is the roofline limiter.
// ---------------------------------------------------------------------------
__global__ __launch_bounds__(256)
void lif_scan_kernel(const float* __restrict__ cur,
                     float* __restrict__ out)
{
    const int    tid     = blockIdx.x * 256 + threadIdx.x;     // 0..262143
    const size_t base    = (size_t)tid * 4;
    const size_t sstride = (size_t)M_DIM * H_DIM;              // 1,048,576

    v4f c   = *(const v4f*)(cur + base);
    v4f mem = {0.f, 0.f, 0.f, 0.f};

    for (int s = 0; s < T_STEPS; ++s) {
        v4f spk;
        #pragma unroll
        for (int e = 0; e < 4; ++e) {
            const float rst = (mem[e] > THRESH) ? THRESH : 0.f;
            mem[e] = BETA * mem[e] + c[e] - rst;
            spk[e] = (mem[e] > THRESH) ? 1.f : 0.f;
        }
        *(v4f*)(out + (size_t)s * sstride + base) = spk;
    }
}

// ---------------------------------------------------------------------------
extern "C" void kernel_launch(void* const* d_in, const int* in_sizes, int n_in,
                              void* d_out, int out_size, void* d_ws, size_t ws_size,
                              hipStream_t stream)
{
    (void)in_sizes; (void)n_in; (void)out_size; (void)ws_size;

    const float* x  = (const float*)d_in[0];   // [32,64,256]
    const float* W1 = (const float*)d_in[1];   // [512,256]
    const float* b1 = (const float*)d_in[2];   // [512]
    float* out = (float*)d_out;                // [32,32,64,512]
    float* cur = (float*)d_ws;                 // 2048*512 floats = 4 MiB scratch

    dim3 g1(M_DIM / 64, H_DIM / 64);           // (32, 8) blocks, 4 waves each
    gemm_cur_kernel<<<g1, 128, 0, stream>>>(x, W1, b1, cur);

    const int nthreads = (M_DIM * H_DIM) / 4;  // 262,144
    lif_scan_kernel<<<nthreads / 256, 256, 0, stream>>>(cur, out);
}